// RoIHeads_69724499083661
// MI455X (gfx1250) — compile-verified
//
#include <hip/hip_runtime.h>
#include <stdint.h>

// Problem shape (fixed by the harness/reference).
#define B_ 16
#define P_ 8000
#define G_ 256
#define FG_THRESH 0.5f
#define BG_THRESH 0.3f

typedef uint32_t u32x4 __attribute__((ext_vector_type(4)));
typedef int      i32x4 __attribute__((ext_vector_type(4)));
typedef int      i32x8 __attribute__((ext_vector_type(8)));

#if defined(__gfx1250__) && __has_builtin(__builtin_amdgcn_tensor_load_to_lds) && \
    __has_builtin(__builtin_amdgcn_s_wait_tensorcnt)
#define HAVE_TDM 1
#else
#define HAVE_TDM 0
#endif

// Low 32 bits of a flat pointer into LDS == byte offset within the WG's LDS
// (ISA 10.2: LDS aperture -> offset is the low 32 address bits).
__device__ __forceinline__ uint32_t lds_offset_of(const void* p) {
  return (uint32_t)(uintptr_t)p;
}

#if HAVE_TDM
// Issue one TDM DMA of `ndw` contiguous DWORDs from global `gsrc` to LDS offset
// `lds_off`. Descriptor per CDNA5 ISA ch.8 (D# groups 0/1; groups 2/3 zero for
// a <=2D tensor). Tracked with TENSORcnt on the issuing wave.
// This toolchain's builtin takes 6 args (g0,g1,g2,g3,g4,cpol).
__device__ __forceinline__ void tdm_load_contig_dwords(const void* gsrc,
                                                       uint32_t lds_off,
                                                       uint32_t ndw) {
  uint64_t ga = (uint64_t)(uintptr_t)gsrc;
  u32x4 g0;
  g0[0] = 1u;                                   // count=1 (valid user descriptor)
  g0[1] = lds_off;                              // lds_addr [63:32]
  g0[2] = (uint32_t)ga;                         // global_addr[31:0]
  g0[3] = (uint32_t)((ga >> 32) & 0x01FFFFFFu)  // global_addr[56:32]
        | (2u << 30);                           // type=2 ("image")
  i32x8 g1;
  g1[0] = (int)(2u << 16);                      // workgroup_mask=0, data_size=2 (4B)
  g1[1] = (int)((ndw & 0xFFFFu) << 16);         // tensor_dim0[15:0]
  g1[2] = (int)(((ndw >> 16) & 0xFFFFu)         // tensor_dim0[31:16]
        | (1u << 16));                          // tensor_dim1 = 1
  g1[3] = (int)((ndw & 0xFFFFu) << 16);         // tile_dim0 = ndw (tensor_dim1 hi = 0)
  g1[4] = 0;                                    // tile_dim1=0, tile_dim2=0 (unused)
  g1[5] = (int)ndw;                             // tensor_dim0_stride[31:0]
  g1[6] = 0;                                    // stride0[47:32]=0, stride1 lo=0
  g1[7] = 0;                                    // stride1 hi = 0
  i32x4 z4 = {0, 0, 0, 0};
  i32x8 z8 = {0, 0, 0, 0, 0, 0, 0, 0};
  __builtin_amdgcn_tensor_load_to_lds(g0, g1, z4, z4, z8, 0);
}
#endif

__global__ __launch_bounds__(256) void roi_match_kernel(
    const float* __restrict__ proposals,   // [B, P, 4]
    const float* __restrict__ gt_boxes,    // [B, G, 4]
    const int*  __restrict__ gt_labels,    // [B, G]
    int* __restrict__ out_idx,             // [B, P]
    int* __restrict__ out_lab)             // [B, P]
{
  __shared__ float4 sbox[G_];   // 4 KB: GT boxes of this batch
  __shared__ int    slab[G_];   // 1 KB: GT labels of this batch

  const int b   = blockIdx.y;
  const int tid = threadIdx.x;
  const int p   = blockIdx.x * blockDim.x + tid;

  const float4* gbox = (const float4*)(gt_boxes + (size_t)b * G_ * 4);
  const int*    glab = gt_labels + (size_t)b * G_;

#if HAVE_TDM
  // One wave (TDM is a per-wave, EXEC-independent DMA) stages the GT tile.
  if (tid < 32) {
    tdm_load_contig_dwords(gbox, lds_offset_of(&sbox[0]), G_ * 4u);
    tdm_load_contig_dwords(glab, lds_offset_of(&slab[0]), G_);
    __builtin_amdgcn_s_wait_tensorcnt(0);   // DMA complete -> LDS populated
  }
  __syncthreads();                           // publish LDS to all 8 waves
#else
  // Cooperative fallback staging.
  sbox[tid] = gbox[tid];
  slab[tid] = glab[tid];
  __syncthreads();
#endif

  float4 pb = make_float4(0.0f, 0.0f, 0.0f, 1.0f);
  if (p < P_) pb = ((const float4*)(proposals + (size_t)b * P_ * 4))[p];
  const float ap = (pb.z - pb.x) * (pb.w - pb.y);

  // Running max/argmax of IoU without per-iteration division:
  //   iou_g > iou_best  <=>  inter_g * union_best > inter_best * union_g
  // (unions strictly positive: box sides >= 1). Exact ties keep the first
  // index, matching jnp.argmax semantics.
  float bi = -1.0f, bu = 1.0f;
  int bidx = 0;
#pragma unroll 4
  for (int g = 0; g < G_; ++g) {
    const float4 gb = sbox[g];               // ds_load_b128 broadcast
    float ltx = fmaxf(gb.x, pb.x);
    float lty = fmaxf(gb.y, pb.y);
    float rbx = fminf(gb.z, pb.z);
    float rby = fminf(gb.w, pb.w);
    float w = fmaxf(rbx - ltx, 0.0f);
    float h = fmaxf(rby - lty, 0.0f);
    float inter = w * h;
    float ag  = (gb.z - gb.x) * (gb.w - gb.y);
    float uni = ag + ap - inter;
    bool better = inter * bu > bi * uni;
    bi   = better ? inter : bi;
    bu   = better ? uni   : bu;
    bidx = better ? g     : bidx;
  }

  if (p < P_) {
    const float val = bi / bu;               // matched max IoU (one IEEE div)
    int idx_out, lab_out;
    if (val < BG_THRESH)      { idx_out = 0;    lab_out = 0;  }  // below -> bg
    else if (val < FG_THRESH) { idx_out = 0;    lab_out = -1; }  // between
    else                      { idx_out = bidx; lab_out = slab[bidx]; }
    const size_t o = (size_t)b * P_ + p;
    out_idx[o] = idx_out;
    out_lab[o] = lab_out;
  }
}

extern "C" void kernel_launch(void* const* d_in, const int* in_sizes, int n_in,
                              void* d_out, int out_size, void* d_ws, size_t ws_size,
                              hipStream_t stream) {
  const float* proposals = (const float*)d_in[0];
  const float* gt_boxes  = (const float*)d_in[1];
  const int*   gt_labels = (const int*)d_in[2];
  int* out_idx = (int*)d_out;                       // first B*P int32 = clamped idx
  int* out_lab = out_idx + (size_t)B_ * P_;         // next  B*P int32 = labels
  dim3 grid((P_ + 255) / 256, B_);
  roi_match_kernel<<<grid, 256, 0, stream>>>(proposals, gt_boxes, gt_labels,
                                             out_idx, out_lab);
}